// AdaptiveLoss_65532611002963
// MI455X (gfx1250) — compile-verified
//
#include <hip/hip_runtime.h>
#include <math.h>

#define NC        17        // classes
#define IGNORE_V  255
#define NSLOT     52        // 17 nominator + 17 sum_comp + 17 sum_p + 1 n_mask
#define NSLOT_PAD 64
#define NBLK      256
#define TPB       256
#define SPATIAL   640000    // 200*200*16  (divisible by 4)
#define NVTOT     1280000   // 2 * SPATIAL
#define NQUAD     320000    // NVTOT / 4

typedef __attribute__((ext_vector_type(2))) float v2f;
typedef __attribute__((ext_vector_type(4))) float v4f;
typedef __attribute__((ext_vector_type(8))) float v8f;
typedef __attribute__((ext_vector_type(4))) int   v4i;

__device__ __forceinline__ float wave_sum32(float v) {
    // deterministic fixed-order wave32 reduction
    v += __shfl_down(v, 16, 32);
    v += __shfl_down(v, 8, 32);
    v += __shfl_down(v, 4, 32);
    v += __shfl_down(v, 2, 32);
    v += __shfl_down(v, 1, 32);
    return v;
}

// Kernel 1: stream pred with b128 loads (4 voxels/thread/iter), per-voxel
// softmax over 17 channels, per-class partial sums per block.
// Slot layout per block (NSLOT_PAD floats in ws):
//   [0..16]  nominator[c]   (sum of p_target where target==c, masked)
//   [17..33] sum_comp[c]    (count of target==c, masked)
//   [34..50] sum_p[c]       (sum of softmax prob of class c over masked voxels)
//   [51]     n_mask
__global__ void __launch_bounds__(TPB)
sem_scal_partials(const float* __restrict__ pred,
                  const int* __restrict__ tgt,
                  float* __restrict__ ws) {
    float accN[NC], accC[NC], accP[NC];
    float accM = 0.0f;
#pragma unroll
    for (int c = 0; c < NC; ++c) { accN[c] = 0.0f; accC[c] = 0.0f; accP[c] = 0.0f; }

    const int qstride = NBLK * TPB;   // 65536 quads per grid stride
    for (int q = blockIdx.x * TPB + threadIdx.x; q < NQUAD; q += qstride) {
        const int g = q * 4;                       // first voxel of the quad
        const int b = (g >= SPATIAL) ? 1 : 0;      // quad never straddles batch
        const int v = g - b * SPATIAL;
        const float* base = pred + (long long)b * NC * SPATIAL + v;

        // hint the next grid-stride tile into cache (speculative, OOB-safe)
        __builtin_prefetch(base + 4 * qstride, 0, 0);

        v4f xv[NC];
#pragma unroll
        for (int c = 0; c < NC; ++c)
            xv[c] = *(const v4f*)(base + (long long)c * SPATIAL);
        const v4i tv = *(const v4i*)(tgt + g);

#pragma unroll
        for (int e = 0; e < 4; ++e) {
            float x[NC];
#pragma unroll
            for (int c = 0; c < NC; ++c) x[c] = xv[c][e];

            float m = x[0];
#pragma unroll
            for (int c = 1; c < NC; ++c) m = fmaxf(m, x[c]);

            float Z = 0.0f;
#pragma unroll
            for (int c = 0; c < NC; ++c) { x[c] = __expf(x[c] - m); Z += x[c]; }
            const float invZ = 1.0f / Z;

            const int t = tv[e];
            if (t != IGNORE_V) {
                accM += 1.0f;
#pragma unroll
                for (int c = 0; c < NC; ++c) {
                    const float p = x[c] * invZ;
                    accP[c] += p;
                    const bool hit = (c == t);
                    accN[c] += hit ? p : 0.0f;
                    accC[c] += hit ? 1.0f : 0.0f;
                }
            }
        }
    }

    // wave32 reduction, then 8-wave LDS reduction (fixed order => deterministic)
#pragma unroll
    for (int c = 0; c < NC; ++c) {
        accN[c] = wave_sum32(accN[c]);
        accC[c] = wave_sum32(accC[c]);
        accP[c] = wave_sum32(accP[c]);
    }
    accM = wave_sum32(accM);

    __shared__ float red[TPB / 32][NSLOT];
    const int lane = threadIdx.x & 31;
    const int wave = threadIdx.x >> 5;
    if (lane == 0) {
#pragma unroll
        for (int c = 0; c < NC; ++c) {
            red[wave][c]          = accN[c];
            red[wave][NC + c]     = accC[c];
            red[wave][2 * NC + c] = accP[c];
        }
        red[wave][NSLOT - 1] = accM;
    }
    __syncthreads();

    if (threadIdx.x < NSLOT_PAD) {
        float s = 0.0f;
        if (threadIdx.x < NSLOT) {
#pragma unroll
            for (int w = 0; w < TPB / 32; ++w) s += red[w][threadIdx.x];
        }
        ws[(long long)blockIdx.x * NSLOT_PAD + threadIdx.x] = s;
    }
}

__device__ __forceinline__ float bce_ones(float x) {
    // F.binary_cross_entropy(x, 1) = min(-log(max(x, 1e-38)), 100)
    return fminf(-__logf(fmaxf(x, 1e-38f)), 100.0f);
}

// Kernel 2: WMMA GEMV reduction  totals = ones(1 x NBLK) * partials(NBLK x 64),
// then the 17-class scalar epilogue. 1 block, 4 waves; wave w owns slots
// [16w, 16w+16). A = ones(16x4) so D[:,n] = sum_k B[k][n] independent of the
// exact K striping of the B operand.
__global__ void __launch_bounds__(128)
sem_scal_finalize(const float* __restrict__ ws,
                  const float* __restrict__ f1_list,
                  float* __restrict__ out) {
    __shared__ float totals[NSLOT_PAD];

    const int lane = threadIdx.x & 31;
    const int wave = threadIdx.x >> 5;     // 0..3
    const int n    = lane & 15;
    const int half = lane >> 4;            // 0 or 1
    const int slot = wave * 16 + n;

    v8f acc = {};
    v2f ones;
    ones[0] = 1.0f;
    ones[1] = 1.0f;

    for (int kb = 0; kb < NBLK; kb += 4) {
        v2f bmat;
        bmat[0] = ws[(long long)(kb + half)     * NSLOT_PAD + slot];
        bmat[1] = ws[(long long)(kb + 2 + half) * NSLOT_PAD + slot];
        acc = __builtin_amdgcn_wmma_f32_16x16x4_f32(
            /*neg_a=*/false, ones, /*neg_b=*/false, bmat,
            /*c_mod=*/(short)0, acc, /*reuse_a=*/false, /*reuse_b=*/false);
    }

    if (lane < 16) totals[slot] = acc[0];  // row M=0 of D holds the column sums
    __syncthreads();

    if (threadIdx.x == 0) {
        const float n_mask = totals[NSLOT - 1];
        float loss_list[NC], new_f1[NC];
        float count = 0.0f;
#pragma unroll
        for (int c = 0; c < NC; ++c) {
            const float nom = totals[c];
            const float cnt = totals[NC + c];
            const float sp  = totals[2 * NC + c];
            const bool  has = cnt > 0.0f;
            const float prec = (sp > 0.0f) ? nom / sp : 0.0f;
            const float rec  = has ? nom / cnt : 0.0f;
            const float negc = n_mask - cnt;
            const float specn = (n_mask - sp) - (cnt - nom);
            const float spec  = (negc > 0.0f) ? specn / negc : 0.0f;

            float ll = 0.0f;
            if (has) {
                ll = bce_ones(rec);
                if (sp > 0.0f)   ll += bce_ones(prec);
                if (negc > 0.0f) ll += bce_ones(spec);
            }
            loss_list[c] = ll;

            const float den = prec + rec;
            const float f1  = (den > 0.0f) ? 2.0f * prec * rec / den : 0.0f;
            const float cur = has ? f1 : 0.0f;
            new_f1[c] = 0.95f * f1_list[c] + 0.05f * cur;
            count += has ? 1.0f : 0.0f;
        }

        const float weight_pool = 3.0f * count;

        // softmax over selected classes (logit = ALPHA*(1-new_f1), else -inf)
        float mx = -INFINITY;
#pragma unroll
        for (int c = 0; c < NC; ++c) {
            if (loss_list[c] != 0.0f) {
                const float lg = 5.0f * (1.0f - new_f1[c]);
                mx = fmaxf(mx, lg);
            }
        }
        float zs = 0.0f;
        float e[NC];
#pragma unroll
        for (int c = 0; c < NC; ++c) {
            if (loss_list[c] != 0.0f) {
                e[c] = __expf(5.0f * (1.0f - new_f1[c]) - mx);
            } else {
                e[c] = 0.0f;
            }
            zs += e[c];
        }
        const float inv_zs = 1.0f / zs;

        float total = 0.0f;
#pragma unroll
        for (int c = 0; c < NC; ++c) {
            total += loss_list[c] * (1.0f + weight_pool * e[c] * inv_zs);
        }
        out[0] = total / (count * 4.0f);   // count * (1 + WPC)
    }
}

extern "C" void kernel_launch(void* const* d_in, const int* in_sizes, int n_in,
                              void* d_out, int out_size, void* d_ws, size_t ws_size,
                              hipStream_t stream) {
    const float* pred    = (const float*)d_in[0];
    const int*   tgt     = (const int*)d_in[1];
    const float* f1_list = (const float*)d_in[2];
    float* out = (float*)d_out;
    float* ws  = (float*)d_ws;   // needs NBLK * NSLOT_PAD * 4 = 64 KB

    sem_scal_partials<<<NBLK, TPB, 0, stream>>>(pred, tgt, ws);
    sem_scal_finalize<<<1, 128, 0, stream>>>(ws, f1_list, out);
}